// RelativePositionEncoder_19628000543013
// MI455X (gfx1250) — compile-verified
//
#include <hip/hip_runtime.h>
#include <stdint.h>

// Table geometry: W rows = [W_pos(66) | W_tok(66) | w_ent(1) | W_ch(6)] x 128 ch
#define TZ      128
#define NROWS   139
#define ROW_TOK 66
#define ROW_ENT 132
#define ROW_CH  133

typedef float v4f __attribute__((ext_vector_type(4)));

__global__ __launch_bounds__(256) void relpos_encode_kernel(
    const int* __restrict__ asym, const int* __restrict__ resi,
    const int* __restrict__ ent,  const int* __restrict__ tok,
    const int* __restrict__ symi, const int* __restrict__ cyc,
    const float* __restrict__ W,  float* __restrict__ out,
    int L, int nTasks)
{
    __shared__ float sW[NROWS * TZ];   // 71168 B of 320 KB WGP LDS

    const int tid = threadIdx.x;

    // ---- Stage W into LDS with CDNA5 async global->LDS copies (ASYNCcnt) ----
    {
        const uint32_t ldsBase = (uint32_t)(uintptr_t)&sW[0]; // low 32b of LDS-aperture ptr = LDS offset
        const int nv4 = (NROWS * TZ) / 4;                     // 4448 x b128
        for (int k = tid; k < nv4; k += 256) {
            uint32_t byteOff = (uint32_t)k << 4;
            uint32_t ldsAddr = ldsBase + byteOff;
            asm volatile("global_load_async_to_lds_b128 %0, %1, %2"
                         :: "v"(ldsAddr), "v"(byteOff), "s"(W) : "memory");
        }
        asm volatile("s_wait_asynccnt 0" ::: "memory");
    }
    __syncthreads();

    const int lane = tid & 31;
    const int c4   = lane << 2;                                   // 4 channels / lane
    const int wv   = __builtin_amdgcn_readfirstlane((int)(tid >> 5)); // uniform wave id

    const v4f we = *(const v4f*)&sW[ROW_ENT * TZ + c4];           // w_ent slice, hoisted

    const int chunks     = L >> 6;                // 64 j's per wave-task
    const int wavesTotal = (int)gridDim.x << 3;   // 8 waves / block

    for (int task = ((int)blockIdx.x << 3) + wv; task < nTasks; task += wavesTotal) {
        const int rowsTot = L * chunks;
        const int bi  = task / rowsTot;
        const int rem = task - bi * rowsTot;
        const int i   = rem / chunks;
        const int j0  = (rem - i * chunks) << 6;

        const int* __restrict__ A = asym + bi * L;
        const int* __restrict__ R = resi + bi * L;
        const int* __restrict__ E = ent  + bi * L;
        const int* __restrict__ T = tok  + bi * L;
        const int* __restrict__ S = symi + bi * L;
        const int* __restrict__ C = cyc  + bi * L;

        const int ai = A[i], ri = R[i], ei = E[i], ti = T[i], si = S[i];
        float* __restrict__ op = out + (((size_t)bi * L + i) * L + j0) * TZ + c4;

        for (int jj = 0; jj < 64; ++jj) {
            const int j  = j0 + jj;
            const int aj = A[j], rj = R[j], ej = E[j], tj = T[j], sj = S[j], cj = C[j];
            const bool same_chain = (ai == aj);

            // rel_pos with cyclic wrap: round-half-even (v_rndne), then int-truncate like .astype
            const int per = (cj > 0) ? cj : 10000;
            int rel = ri - rj;
            rel -= (int)((float)per * rintf((float)rel / (float)per));

            int dres = rel + 32;        dres = dres < 0 ? 0 : (dres > 64 ? 64 : dres);
            if (!same_chain) dres = 65;

            int dtok = ti - tj + 32;    dtok = dtok < 0 ? 0 : (dtok > 64 ? 64 : dtok);
            if (!(same_chain && (ri == rj))) dtok = 65;

            int dch = si - sj + 2;      dch = dch < 0 ? 0 : (dch > 4 ? 4 : dch);
            if (same_chain) dch = 5;

            const float se = (ei == ej) ? 1.0f : 0.0f;

            // three conflict-free ds_load_b128 gathers (row uniform, lanes span 512B)
            const v4f p0 = *(const v4f*)&sW[dres * TZ + c4];
            const v4f p1 = *(const v4f*)&sW[(ROW_TOK + dtok) * TZ + c4];
            const v4f p2 = *(const v4f*)&sW[(ROW_CH  + dch ) * TZ + c4];

            v4f o;
            o.x = fmaf(se, we.x, p0.x + p1.x + p2.x);
            o.y = fmaf(se, we.y, p0.y + p1.y + p2.y);
            o.z = fmaf(se, we.z, p0.z + p1.z + p2.z);
            o.w = fmaf(se, we.w, p0.w + p1.w + p2.w);

            // streaming store: wave writes one contiguous 512B slab, NT (output >> L2)
            __builtin_nontemporal_store(o, (v4f*)(op + (size_t)jj * TZ));
        }
    }
}

extern "C" void kernel_launch(void* const* d_in, const int* in_sizes, int n_in,
                              void* d_out, int out_size, void* d_ws, size_t ws_size,
                              hipStream_t stream) {
    const int*   asym = (const int*)d_in[0];
    const int*   resi = (const int*)d_in[1];
    const int*   ent  = (const int*)d_in[2];
    const int*   tok  = (const int*)d_in[3];
    const int*   symi = (const int*)d_in[4];
    const int*   cyc  = (const int*)d_in[5];
    const float* W    = (const float*)d_in[6];
    float*       out  = (float*)d_out;

    const int BL = in_sizes[0];                         // B*L
    const int L  = (int)((long long)out_size / ((long long)TZ * BL)); // out = B*L*L*TZ
    const int B  = BL / L;
    const int chunks = L / 64;
    const int nTasks = B * L * chunks;                  // 16384 for B=1,L=1024

    int blocks = (nTasks + 7) / 8;                      // 8 waves per block, 1 task/wave
    if (blocks > 4096) blocks = 4096;

    hipLaunchKernelGGL(relpos_encode_kernel, dim3(blocks), dim3(256), 0, stream,
                       asym, resi, ent, tok, symi, cyc, W, out, L, nTasks);
}